// ViConstituentModule_69844758167799
// MI455X (gfx1250) — compile-verified
//
#include <hip/hip_runtime.h>
#include <hip/hip_bf16.h>

#define HID 768
#define NHEAD 12
#define HDIM 64
#define BB 16
#define MM 512
#define NM (BB * MM)          // 8192 rows of q
#define NPAIR (BB * NHEAD)    // 192 (b,h) pairs
#define EPS 1.0e-4f
#define LOG_GUARD 1.0e-9f

typedef float v2f __attribute__((ext_vector_type(2)));
typedef float v4f __attribute__((ext_vector_type(4)));
typedef float v8f __attribute__((ext_vector_type(8)));

// -------------------------------------------------------------------------
// Kernel 1: q = X @ Wq + bq   (f32 GEMM via V_WMMA_F32_16X16X4_F32)
// X: (8192, 768) row-major, W: (768, 768) row-major, q: (8192, 768).
// One wave32 computes one 16x16 tile, K-loop of 768 in steps of 4.
// A layout (16x4 f32): lane = {hi=k-half, lo=m}; a.x=k(2*hi), a.y=k(2*hi+1)
// B layout (4x16 f32): b.x=B[2*hi][lo], b.y=B[2*hi+1][lo]
// C/D layout: vgpr r -> M = r + 8*hi, N = lo
// -------------------------------------------------------------------------
__global__ __launch_bounds__(256)
void vicm_qgemm_wmma(const float* __restrict__ X, const float* __restrict__ W,
                     const float* __restrict__ bq, float* __restrict__ q) {
    const int lane = threadIdx.x & 31;
    const int wave = threadIdx.x >> 5;
    const int w = blockIdx.x * 8 + wave;           // tile id
    const int TILES_N = HID / 16;                  // 48
    if (w >= (NM / 16) * TILES_N) return;
    const int tn = w % TILES_N;
    const int tm = w / TILES_N;
    const int row0 = tm * 16;
    const int col0 = tn * 16;
    const int lo = lane & 15;
    const int hi = lane >> 4;

    v8f acc = {};
    const float* xrow = X + (size_t)(row0 + lo) * HID;   // A: row m = lo
    #pragma unroll 4
    for (int k = 0; k < HID; k += 4) {
        v2f a = *(const v2f*)(xrow + k + 2 * hi);
        v2f b;
        b.x = W[(size_t)(k + 2 * hi) * HID + col0 + lo];
        b.y = W[(size_t)(k + 2 * hi + 1) * HID + col0 + lo];
        acc = __builtin_amdgcn_wmma_f32_16x16x4_f32(
            false, a, false, b, (short)0, acc, false, false);
    }
    const float bias = bq[col0 + lo];
    #pragma unroll
    for (int r = 0; r < 8; ++r) {
        const int m = r + 8 * hi;
        q[(size_t)(row0 + m) * HID + col0 + lo] = acc[r] + bias;
    }
}

// -------------------------------------------------------------------------
// Kernel 2: per (b,h): neighbor dots -> 2-entry softmax -> s_i -> prefix sums
// grid = (NHEAD, BB), block = 512
// outputs: Sws[pair][k] = sum_{j<k} log(na[j,j+1]+1e-9);  sbws[pair][i] = s_i
// -------------------------------------------------------------------------
__global__ __launch_bounds__(512)
void vicm_bands(const float* __restrict__ q, const float* __restrict__ prior,
                const int* __restrict__ mask,
                float* __restrict__ Sws, float* __restrict__ sbws) {
    __shared__ float sh_d[MM];
    __shared__ float sh_aL[MM];
    __shared__ float sh_aR[MM];
    __shared__ float sh_l[MM];
    __shared__ float sh_s[MM];

    const int h = blockIdx.x;
    const int b = blockIdx.y;
    const int i = threadIdx.x;

    // phase 1: superdiagonal scores d_i = q_i . q_{i+1} / 64
    if (i < MM - 1) {
        const float* qa = q + ((size_t)(b * MM + i) * HID + h * HDIM);
        const float* qb = qa + HID;
        float s = 0.f;
        #pragma unroll
        for (int t = 0; t < HDIM; ++t) s += qa[t] * qb[t];
        sh_d[i] = s * (1.0f / HDIM);
    } else {
        sh_d[i] = 0.f;
    }
    __syncthreads();

    // phase 2: per-row softmax over the (at most 2) unmasked neighbors
    {
        const int r = i;
        const bool hasL = (r >= 1) && (mask[b * MM + r - 1] != 0);
        const bool hasR = (r <= MM - 2) && (mask[b * MM + r + 1] != 0);
        const float dL = (r >= 1) ? sh_d[r - 1] : 0.f;
        const float dR = sh_d[r];
        float aL = 0.f, aR = 0.f;
        if (hasL && hasR) {
            const float m = fmaxf(dL, dR);
            const float eL = expf(dL - m), eR = expf(dR - m);
            const float z = eL + eR;
            aL = eL / z; aR = eR / z;
        } else if (hasL) {
            aL = 1.f;
        } else if (hasR) {
            aR = 1.f;
        }
        sh_aL[r] = aL;
        sh_aR[r] = aR;
    }
    __syncthreads();

    // phase 3: symmetrized band value + log(c_i)
    if (i < MM - 1) {
        const float s = sqrtf(sh_aR[i] * sh_aL[i + 1] + EPS);
        sh_s[i] = s;
        const float pv = prior[(size_t)b * MM * MM + (size_t)i * MM + (i + 1)];
        const float c = pv + (1.0f - pv) * s + LOG_GUARD;
        sh_l[i] = logf(c);
    } else {
        sh_s[i] = 0.f;
        sh_l[i] = 0.f;
    }
    __syncthreads();

    // phase 4: exclusive scan S_k = sum_{j<k} l_j  (Hillis-Steele in LDS)
    __shared__ float sc[MM];
    sc[i] = (i == 0) ? 0.f : sh_l[i - 1];
    __syncthreads();
    for (int off = 1; off < MM; off <<= 1) {
        float v = sc[i];
        if (i >= off) v += sc[i - off];
        __syncthreads();
        sc[i] = v;
        __syncthreads();
    }

    const size_t base = (size_t)(b * NHEAD + h) * MM;
    Sws[base + i]  = sc[i];
    sbws[base + i] = sh_s[i];
}

// -------------------------------------------------------------------------
// Kernel 3: write g_attn and neibor_attn (the 402 MB streaming part)
// grid = (NHEAD, 64, BB): x fastest so all 12 heads reuse the prior tile in L2
// block = 256, tile = 64(i) x 64(j), float4 per thread per row, NT stores
// -------------------------------------------------------------------------
__global__ __launch_bounds__(256)
void vicm_write(const float* __restrict__ prior,
                const float* __restrict__ Sws, const float* __restrict__ sbws,
                float* __restrict__ out_g, float* __restrict__ out_na) {
    const int h = blockIdx.x;
    const int b = blockIdx.z;
    const int it = blockIdx.y >> 3;
    const int jt = blockIdx.y & 7;
    const int i0 = it * 64;
    const int j0 = jt * 64;

    const int tid = threadIdx.x;
    const int jq = tid & 15;
    const int r0 = tid >> 4;
    const int j = j0 + jq * 4;

    const size_t pair = (size_t)(b * NHEAD + h);
    const float* S  = Sws  + pair * MM;
    const float* sb = sbws + pair * MM;

    const v4f Sj  = *(const v4f*)(S + j);
    const v4f sbj = *(const v4f*)(sb + j);
    const float base0 = sqrtf(EPS);   // 0.01 (bit-matched to reference)

    #pragma unroll
    for (int s = 0; s < 4; ++s) {
        const int i = i0 + r0 + 16 * s;
        const float Si  = S[i];
        const float sbi = sb[i];
        const v4f p4 = *(const v4f*)(prior + (size_t)b * MM * MM + (size_t)i * MM + j);

        v4f na4, g4;
        #pragma unroll
        for (int c = 0; c < 4; ++c) {
            const int jj = j + c;
            const float p = p4[c];
            float base = base0;
            if (jj == i + 1)      base = sbi;
            else if (jj == i - 1) base = sbj[c];
            const float na = p + (1.0f - p) * base;
            float g;
            if (jj == i) {
                g = na;
            } else {
                const float dS = (jj > i) ? (Sj[c] - Si) : (Si - Sj[c]);
                g = __expf(dS) + EPS;
            }
            na4[c] = na;
            g4[c]  = g;
        }
        const size_t off = (pair * MM + (size_t)i) * MM + j;
        __builtin_nontemporal_store(g4,  (v4f*)(out_g  + off));
        __builtin_nontemporal_store(na4, (v4f*)(out_na + off));
    }
}

extern "C" void kernel_launch(void* const* d_in, const int* in_sizes, int n_in,
                              void* d_out, int out_size, void* d_ws, size_t ws_size,
                              hipStream_t stream) {
    const float* ocr   = (const float*)d_in[0];   // (16, 512, 768)
    const float* prior = (const float*)d_in[1];   // (16, 1, 512, 512)
    const float* Wq    = (const float*)d_in[2];   // (768, 768)
    const float* bq    = (const float*)d_in[3];   // (768)
    const int*   msk   = (const int*)d_in[4];     // (16, 512)

    float* qws  = (float*)d_ws;                   // 8192*768 floats (25.2 MB)
    float* Sws  = qws + (size_t)NM * HID;         // 192*512 floats
    float* sbws = Sws + (size_t)NPAIR * MM;       // 192*512 floats

    float* out_g  = (float*)d_out;                          // (16,12,512,512)
    float* out_na = out_g + (size_t)BB * NHEAD * MM * MM;   // (16,12,512,512)

    // 1) q = ocr @ Wq + bq  (WMMA f32): 24576 tiles, 8 waves/block
    vicm_qgemm_wmma<<<dim3(3072), dim3(256), 0, stream>>>(ocr, Wq, bq, qws);

    // 2) per-(b,h) band values + log prefix sums
    vicm_bands<<<dim3(NHEAD, BB), dim3(512), 0, stream>>>(qws, prior, msk, Sws, sbws);

    // 3) stream the two (B,H,M,M) outputs
    vicm_write<<<dim3(NHEAD, 64, BB), dim3(256), 0, stream>>>(prior, Sws, sbws, out_g, out_na);
}